// VQCLayer_85796266705041
// MI455X (gfx1250) — compile-verified
//
#include <hip/hip_runtime.h>

#define NQ       12
#define DIM      4096          // 2^12 amplitudes
#define NLAYERS  6
#define TPB      256           // 8 wave32s per block
#define APT      (DIM / TPB)   // 16 amplitudes per thread
#define ANG_STRIDE 16          // padded angles row (12 used)
#define DOUT     256

typedef float v2f __attribute__((ext_vector_type(2)));
typedef float v8f __attribute__((ext_vector_type(8)));

// ---------------------------------------------------------------------------
// Kernel 1: angles[4096,16] = x[4096,256] @ W_in^T (+ b_in), fp32 WMMA.
// One wave per 16-row tile. 8 K-values per iteration via one b128 load per
// operand feeding two V_WMMA_F32_16X16X4_F32 (K order is commutative, so
// abs-K -> rel-K remapping per lane-half is exact).
// A-frag (16x4 f32): lanes 0-15 -> rel K{0,1}, lanes 16-31 -> rel K{2,3}.
// B-frag (4x16 f32): same half split; N = lane&15.
// C/D (16x16 f32, 8 VGPRs): VGPR r, lanes 0-15 -> M=r, lanes 16-31 -> M=r+8.
// Invalid columns (n >= 12) handled branch-free: clamped row + 0.0 mask.
// ---------------------------------------------------------------------------
__global__ __launch_bounds__(256) void fc_in_wmma(const float* __restrict__ x,
                                                  const float* __restrict__ W_in,
                                                  const float* __restrict__ b_in,
                                                  float* __restrict__ angles) {
    const int lane = threadIdx.x & 31;
    const int wave = threadIdx.x >> 5;
    const int tile = blockIdx.x * 8 + wave;      // 256 tiles of 16 rows
    const int n    = lane & 15;                  // output col (0..15; 12 valid)
    const int h    = lane >> 4;                  // lane half: rel K{0,1} vs {2,3}
    const int nc   = (n < NQ) ? n : (NQ - 1);    // clamped W row (branch-free)
    const float mask = (n < NQ) ? 1.0f : 0.0f;

    const float* xrow = x + (size_t)(tile * 16 + n) * 256;  // A row = lane&15
    const float* wrow = W_in + nc * 256;

    v8f acc = {};
    for (int k0 = 0; k0 < 256; k0 += 8) {
        const float4 av = *(const float4*)(xrow + k0 + 4 * h);
        float4 wv       = *(const float4*)(wrow + k0 + 4 * h);
        wv.x *= mask; wv.y *= mask; wv.z *= mask; wv.w *= mask;

        v2f a1, b1, a2, b2;
        a1.x = av.x; a1.y = av.y;  b1.x = wv.x; b1.y = wv.y;
        a2.x = av.z; a2.y = av.w;  b2.x = wv.z; b2.y = wv.w;
        // WMMA1: abs K {k0, k0+1, k0+4, k0+5}
        acc = __builtin_amdgcn_wmma_f32_16x16x4_f32(false, a1, false, b1,
                                                    (short)0, acc, false, false);
        // WMMA2: abs K {k0+2, k0+3, k0+6, k0+7}
        acc = __builtin_amdgcn_wmma_f32_16x16x4_f32(false, a2, false, b2,
                                                    (short)0, acc, false, false);
    }

    const float bias = b_in[nc] * mask;
    const int mbase = tile * 16 + (h << 3);
#pragma unroll
    for (int r = 0; r < 8; ++r)
        angles[(size_t)(mbase + r) * ANG_STRIDE + n] = acc[r] + bias;
}

// ---------------------------------------------------------------------------
// Kernel 2: fused state-vector sim + Z expvals + fc_out.
// One block per sample; 32 KB AoS float2 state resident in LDS
// (ds_load_b64/ds_store_b64, conflict-free for large-stride butterflies).
// ---------------------------------------------------------------------------
__device__ __forceinline__ void rx_apply(float2* st, int bit, float c, float s) {
    const int tid = threadIdx.x;
#pragma unroll
    for (int it = 0; it < (DIM / 2) / TPB; ++it) {   // 8 pairs per thread
        int p  = tid + it * TPB;
        int i0 = ((p >> bit) << (bit + 1)) | (p & ((1 << bit) - 1));
        int i1 = i0 | (1 << bit);
        float2 a0 = st[i0];
        float2 a1 = st[i1];
        // RX: a0' = c*a0 - i s*a1 ; a1' = -i s*a0 + c*a1
        float2 r0, r1;
        r0.x = fmaf(c, a0.x,  s * a1.y);
        r0.y = fmaf(c, a0.y, -s * a1.x);
        r1.x = fmaf(c, a1.x,  s * a0.y);
        r1.y = fmaf(c, a1.y, -s * a0.x);
        st[i0] = r0;
        st[i1] = r1;
    }
    __syncthreads();
}

__device__ __forceinline__ void cnot_apply(float2* st, int cbit, int tbit) {
    const int tid = threadIdx.x;
    const int lo = (cbit < tbit) ? cbit : tbit;
    const int hi = (cbit < tbit) ? tbit : cbit;
#pragma unroll
    for (int it = 0; it < (DIM / 4) / TPB; ++it) {   // 4 swaps per thread
        int p = tid + it * TPB;
        int i = ((p >> lo) << (lo + 1)) | (p & ((1 << lo) - 1));
        i     = ((i >> hi) << (hi + 1)) | (i & ((1 << hi) - 1));
        i    |= (1 << cbit);            // control = 1, target = 0
        int j = i | (1 << tbit);        // target flipped
        float2 t = st[i];
        st[i] = st[j];
        st[j] = t;
    }
    __syncthreads();
}

__global__ __launch_bounds__(TPB) void vqc_kernel(const float* __restrict__ angles,
                                                  const float* __restrict__ qp,
                                                  const float* __restrict__ W_out,
                                                  const float* __restrict__ b_out,
                                                  float* __restrict__ out) {
    __shared__ float2 st[DIM];       // 32 KB state
    const int tid = threadIdx.x;
    const int b   = blockIdx.x;

    // |0...0>
#pragma unroll
    for (int j = 0; j < APT; ++j) {
        float2 z; z.x = 0.f; z.y = 0.f;
        st[tid * APT + j] = z;
    }
    if (tid == 0) { float2 one; one.x = 1.f; one.y = 0.f; st[0] = one; }
    __syncthreads();

    // data-encoding RX layer (per-sample angles); qubit q lives on bit (11-q)
    for (int q = 0; q < NQ; ++q) {
        float hh = 0.5f * angles[(size_t)b * ANG_STRIDE + q];
        rx_apply(st, 11 - q, __cosf(hh), __sinf(hh));
    }

    // variational layers: RX (shared weights) + CNOT ring
    for (int l = 0; l < NLAYERS; ++l) {
        for (int q = 0; q < NQ; ++q) {
            float hh = 0.5f * qp[l * NQ + q];
            rx_apply(st, 11 - q, __cosf(hh), __sinf(hh));
        }
        for (int q = 0; q < NQ - 1; ++q)
            cnot_apply(st, 11 - q, 10 - q);   // CNOT(q, q+1)
        cnot_apply(st, 0, 11);                // CNOT(11, 0) ring closure
    }

    // Z expvals: per-thread signed-probability partials
    float zp[NQ];
#pragma unroll
    for (int q = 0; q < NQ; ++q) zp[q] = 0.f;
#pragma unroll
    for (int j = 0; j < APT; ++j) {
        int i = tid * APT + j;
        float2 a = st[i];
        float pr = fmaf(a.x, a.x, a.y * a.y);
#pragma unroll
        for (int q = 0; q < NQ; ++q)
            zp[q] += ((i >> (11 - q)) & 1) ? -pr : pr;
    }
    __syncthreads();                 // done reading state; reuse LDS as scratch
    float* red = (float*)st;         // 256*12 floats fits in 8192-float region
#pragma unroll
    for (int q = 0; q < NQ; ++q) red[tid * NQ + q] = zp[q];
    __syncthreads();
    for (int s = TPB / 2; s >= 1; s >>= 1) {
        if (tid < s) {
#pragma unroll
            for (int q = 0; q < NQ; ++q)
                red[tid * NQ + q] += red[(tid + s) * NQ + q];
        }
        __syncthreads();
    }

    // fused fc_out: out[b][tid] = z . W_out[tid,:] + b_out[tid]
    float accv = b_out[tid];
#pragma unroll
    for (int q = 0; q < NQ; ++q)
        accv = fmaf(red[q], W_out[tid * NQ + q], accv);
    out[(size_t)b * DOUT + tid] = accv;
}

// ---------------------------------------------------------------------------
extern "C" void kernel_launch(void* const* d_in, const int* in_sizes, int n_in,
                              void* d_out, int out_size, void* d_ws, size_t ws_size,
                              hipStream_t stream) {
    const float* x        = (const float*)d_in[0];  // [4096,256]
    const float* W_in     = (const float*)d_in[1];  // [12,256]
    const float* b_in     = (const float*)d_in[2];  // [12]
    const float* q_params = (const float*)d_in[3];  // [6,12]
    const float* W_out    = (const float*)d_in[4];  // [256,12]
    const float* b_out    = (const float*)d_in[5];  // [256]
    float* out    = (float*)d_out;                  // [4096,256]
    float* angles = (float*)d_ws;                   // [4096,16] padded, 256 KB

    fc_in_wmma<<<dim3(32),   dim3(256), 0, stream>>>(x, W_in, b_in, angles);
    vqc_kernel<<<dim3(4096), dim3(TPB), 0, stream>>>(angles, q_params, W_out, b_out, out);
}